// TGAT_58892591562894
// MI455X (gfx1250) — compile-verified
//
#include <hip/hip_runtime.h>
#include <math.h>

typedef float v2f __attribute__((ext_vector_type(2)));
typedef float v8f __attribute__((ext_vector_type(8)));

#define LDSTRIDE 372   // 368 cols + pad; stride%64=52, gcd(52,64)=4 -> conflict-free A reads
#define ROWS     16
#define NWAVES   8

// ---- workspace layout (float offsets) -------------------------------------
#define OFF_C1   0                      // c1[356]
#define OFF_C2   384                    // c2[368]  (zero-padded bias for stage A')
#define OFF_BLP  768                    // blp[256] (lp_src_b + lp_dst_b)
#define OFF_P1   1024                   // packed fused (Wv'@Wo): 23 tiles x 43 ksteps x 64
#define SZ_P1    (23 * 43 * 64)
#define OFF_P2   (OFF_P1 + SZ_P1)       // packed fc1:  16 x 92 x 64
#define SZ_P2    (16 * 92 * 64)
#define OFF_P3   (OFF_P2 + SZ_P2)       // packed fc2:  16 x 64 x 64
#define SZ_P3    (16 * 64 * 64)
#define OFF_P4   (OFF_P3 + SZ_P3)       // packed lp_src_w+lp_dst_w: 16 x 64 x 64
#define SZ_P4    (16 * 64 * 64)

// ---------------------------------------------------------------------------
// Setup 1: c1[j] = bv[j] + sum_t cos(t2v_b[t]) * Wv[(428+t)*356 + j]
// ---------------------------------------------------------------------------
__global__ void k_c1(const float* __restrict__ t2vb, const float* __restrict__ Wv,
                     const float* __restrict__ bv, float* __restrict__ ws) {
    int j = blockIdx.x * blockDim.x + threadIdx.x;
    if (j >= 356) return;
    float acc = bv[j];
    for (int t = 0; t < 100; ++t)
        acc += cosf(t2vb[t]) * Wv[(428 + t) * 356 + j];
    ws[OFF_C1 + j] = acc;
}

// Setup 2: c2[j] = bo[j] + sum_k c1[k]*Wo[k*356+j]   (j<356), zero pad to 368
__global__ void k_c2(const float* __restrict__ Wo, const float* __restrict__ bo,
                     float* __restrict__ ws) {
    int j = blockIdx.x * blockDim.x + threadIdx.x;
    if (j >= 368) return;
    float acc = 0.f;
    if (j < 356) {
        acc = bo[j];
        for (int k = 0; k < 356; ++k)
            acc += ws[OFF_C1 + k] * Wo[k * 356 + j];
    }
    ws[OFF_C2 + j] = acc;
}

// Setup 3: blp = lp_src_b + lp_dst_b
__global__ void k_blp(const float* __restrict__ a, const float* __restrict__ b,
                      float* __restrict__ ws) {
    int j = threadIdx.x;
    ws[OFF_BLP + j] = a[j] + b[j];
}

// Packed B-fragment index -> (kr, nc):
//   idx = (t*(Kpad/4) + s)*64 + lane*2 + e
//   kr = 4s + 2*(lane>>4) + e ;  nc = 16t + (lane&15)
__device__ __forceinline__ void unpack_idx(int idx, int ksteps, int& t, int& kr, int& nc) {
    int per = ksteps * 64;
    t = idx / per;
    int rem = idx - t * per;
    int s = rem >> 6, l2 = rem & 63;
    int lane = l2 >> 1, e = l2 & 1;
    kr = 4 * s + ((lane >> 4) << 1) + e;
    nc = 16 * t + (lane & 15);
}

// Setup 4: packed fused A-stage weight  P1[kr][nc] = sum_j Wv'[kr][j]*Wo[j][nc]
__global__ void k_pack_fused(const float* __restrict__ WvE, const float* __restrict__ Wo,
                             float* __restrict__ ws) {
    int idx = blockIdx.x * blockDim.x + threadIdx.x;
    if (idx >= SZ_P1) return;
    int t, kr, nc; unpack_idx(idx, 43, t, kr, nc);
    float acc = 0.f;
    if (nc < 356) {                                  // kr < 172 always (Kpad==Kact)
        for (int j = 0; j < 356; ++j)
            acc += WvE[kr * 356 + j] * Wo[j * 356 + nc];
    }
    ws[OFF_P1 + idx] = acc;
}

// Setup 5: generic packer (optionally summing two weights), ldW == N == 256 here
__global__ void k_pack(const float* __restrict__ W, const float* __restrict__ W2,
                       float* __restrict__ dst, int ksteps, int Kact, int N, int total) {
    int idx = blockIdx.x * blockDim.x + threadIdx.x;
    if (idx >= total) return;
    int t, kr, nc; unpack_idx(idx, ksteps, t, kr, nc);
    float v = 0.f;
    if (kr < Kact && nc < N) {
        v = W[kr * N + nc];
        if (W2) v += W2[kr * N + nc];
    }
    dst[idx] = v;
}

// ---------------------------------------------------------------------------
// One 16-row GEMM stage: branch-free ds_load_b64 + global_load_b64 + wmma.
// ---------------------------------------------------------------------------
template<int KSTEPS, bool RELU>
__device__ __forceinline__ void gemm16(
    const float* sIn, float* sOut,
    const float* __restrict__ Wp, const float* __restrict__ bias,
    int nTiles, int wave, int lane)
{
    const int mA    = lane & 15;
    const int kHi   = (lane >> 4) << 1;
    const int nB    = lane & 15;
    const int mBase = (lane >> 4) << 3;
    const float* sA = sIn + mA * LDSTRIDE + kHi;

    for (int t = wave; t < nTiles; t += NWAVES) {
        const float* wp = Wp + (size_t)t * (KSTEPS * 64) + lane * 2;
        v8f acc = {0.f, 0.f, 0.f, 0.f, 0.f, 0.f, 0.f, 0.f};
        #pragma unroll
        for (int s = 0; s < KSTEPS; ++s) {
            v2f a = *(const v2f*)&sA[4 * s];
            v2f b = *(const v2f*)&wp[s * 64];
            acc = __builtin_amdgcn_wmma_f32_16x16x4_f32(
                false, a, false, b, (short)0, acc, false, false);
        }
        const float bval = bias[t * 16 + nB];       // pad cols carry zero bias
        #pragma unroll
        for (int i = 0; i < 8; ++i) {
            float v = acc[i] + bval;
            if (RELU) v = fmaxf(v, 0.f);
            sOut[(mBase + i) * LDSTRIDE + t * 16 + nB] = v;
        }
    }
}

// ---------------------------------------------------------------------------
// Fused per-16-row pipeline: E -> X2 -> H -> Z -> HP -> sigmoid
// ---------------------------------------------------------------------------
__global__ __launch_bounds__(256) void tgat_fused_kernel(
    const float* __restrict__ edge, const float* __restrict__ ws,
    const float* __restrict__ fc1b, const float* __restrict__ fc2b,
    const float* __restrict__ lpow, const float* __restrict__ lpob,
    float* __restrict__ out, int B)
{
    __shared__ float sPing[ROWS * LDSTRIDE];
    __shared__ float sPong[ROWS * LDSTRIDE];
    __shared__ float sRed[ROWS * 16];

    const int tid  = threadIdx.x;
    const int wave = tid >> 5;
    const int lane = tid & 31;
    const int row0 = blockIdx.x * ROWS;

    // load edge_feats tile [16 x 172] into pong
    for (int idx = tid; idx < ROWS * 172; idx += 256) {
        int r = idx / 172, c = idx - r * 172;
        int gr = row0 + r;
        sPong[r * LDSTRIDE + c] = (gr < B) ? edge[gr * 172 + c] : 0.f;
    }
    __syncthreads();

    // A': X2 = E @ (Wv'@Wo) + c2          (K=172, 23 n-tiles, pads are zero)
    gemm16<43, false>(sPong, sPing, ws + OFF_P1, ws + OFF_C2, 23, wave, lane);
    __syncthreads();
    // C: H = relu(X2 @ fc1w + fc1b)       (K=356->368, 16 n-tiles)
    gemm16<92, true>(sPing, sPong, ws + OFF_P2, fc1b, 16, wave, lane);
    __syncthreads();
    // D: Z = H @ fc2w + fc2b              (256x256)
    gemm16<64, false>(sPong, sPing, ws + OFF_P3, fc2b, 16, wave, lane);
    __syncthreads();
    // E: HP = relu(Z @ (lps+lpd) + blp)   (256x256)
    gemm16<64, true>(sPing, sPong, ws + OFF_P4, ws + OFF_BLP, 16, wave, lane);
    __syncthreads();

    // F: prob = sigmoid(HP @ lp_out_w + lp_out_b); pos == neg == prob
    {
        int r = tid >> 4, c = tid & 15;
        float p = 0.f;
        #pragma unroll
        for (int j = 0; j < 16; ++j)
            p += sPong[r * LDSTRIDE + c * 16 + j] * lpow[c * 16 + j];
        sRed[r * 16 + c] = p;
    }
    __syncthreads();
    if (tid < ROWS) {
        float s = lpob[0];
        #pragma unroll
        for (int j = 0; j < 16; ++j) s += sRed[tid * 16 + j];
        float prob = 1.f / (1.f + expf(-s));
        int gr = row0 + tid;
        if (gr < B) { out[gr] = prob; out[B + gr] = prob; }
    }
}

// ---------------------------------------------------------------------------
extern "C" void kernel_launch(void* const* d_in, const int* in_sizes, int n_in,
                              void* d_out, int out_size, void* d_ws, size_t ws_size,
                              hipStream_t stream) {
    // 0 src 1 dst 2 time 3 edge_feats 4 t2v_w 5 t2v_b 6 Wq 7 bq 8 Wk 9 bk
    // 10 Wv 11 bv 12 Wo 13 bo 14 fc1_w 15 fc1_b 16 fc2_w 17 fc2_b
    // 18 lp_src_w 19 lp_src_b 20 lp_dst_w 21 lp_dst_b 22 lp_out_w 23 lp_out_b
    const float* edge = (const float*)d_in[3];
    const float* t2vb = (const float*)d_in[5];
    const float* Wv   = (const float*)d_in[10];
    const float* bv   = (const float*)d_in[11];
    const float* Wo   = (const float*)d_in[12];
    const float* bo   = (const float*)d_in[13];
    const float* fc1w = (const float*)d_in[14];
    const float* fc1b = (const float*)d_in[15];
    const float* fc2w = (const float*)d_in[16];
    const float* fc2b = (const float*)d_in[17];
    const float* lpsw = (const float*)d_in[18];
    const float* lpsb = (const float*)d_in[19];
    const float* lpdw = (const float*)d_in[20];
    const float* lpdb = (const float*)d_in[21];
    const float* lpow = (const float*)d_in[22];
    const float* lpob = (const float*)d_in[23];
    float* out = (float*)d_out;
    float* ws  = (float*)d_ws;          // ~1.16 MB used
    const int B = in_sizes[0];

    k_c1 <<<2, 192, 0, stream>>>(t2vb, Wv, bv, ws);
    k_c2 <<<2, 192, 0, stream>>>(Wo, bo, ws);
    k_blp<<<1, 256, 0, stream>>>(lpsb, lpdb, ws);
    k_pack_fused<<<(SZ_P1 + 255) / 256, 256, 0, stream>>>(Wv + 256 * 356, Wo, ws);
    k_pack<<<(SZ_P2 + 255) / 256, 256, 0, stream>>>(fc1w, nullptr, ws + OFF_P2, 92, 356, 256, SZ_P2);
    k_pack<<<(SZ_P3 + 255) / 256, 256, 0, stream>>>(fc2w, nullptr, ws + OFF_P3, 64, 256, 256, SZ_P3);
    k_pack<<<(SZ_P4 + 255) / 256, 256, 0, stream>>>(lpsw, lpdw,   ws + OFF_P4, 64, 256, 256, SZ_P4);

    const int grid = (B + ROWS - 1) / ROWS;
    tgat_fused_kernel<<<grid, 256, 0, stream>>>(edge, ws, fc1b, fc2b, lpow, lpob, out, B);
}